// SleepyDyHead_23983097381017
// MI455X (gfx1250) — compile-verified
//
#include <hip/hip_runtime.h>
#include <math.h>

#define CCH   256
#define RDIM  64
#define KCH   72          // K-chunk: 8 input channels x 9 taps

typedef __attribute__((ext_vector_type(2))) float v2f;
typedef __attribute__((ext_vector_type(8))) float v8f;

__device__ __forceinline__ float hsig(float x) {
    float v = x * (1.0f / 6.0f) + 0.5f;
    return fminf(fmaxf(v, 0.0f), 1.0f);
}

// ---------------------------------------------------------------------------
// Offset conv: 3x3, pad 1, C->27.  Channels >=18 get sigmoid (mask).
// out layout: [27, H*W]
// ---------------------------------------------------------------------------
__global__ void off_conv_kernel(const float* __restrict__ x,
                                const float* __restrict__ w,
                                const float* __restrict__ b,
                                float* __restrict__ out, int H, int W) {
    int idx = blockIdx.x * blockDim.x + threadIdx.x;
    int HW = H * W;
    if (idx >= 27 * HW) return;
    int hw = idx % HW;
    int o  = idx / HW;
    int yy = hw / W, xx = hw % W;
    float acc = b[o];
    const float* wo = w + (size_t)o * CCH * 9;
    for (int c = 0; c < CCH; ++c) {
        const float* xc = x + (size_t)c * HW;
        const float* wc = wo + c * 9;
        #pragma unroll
        for (int ky = 0; ky < 3; ++ky) {
            int iy = yy + ky - 1;
            if (iy < 0 || iy >= H) continue;
            #pragma unroll
            for (int kx = 0; kx < 3; ++kx) {
                int ix = xx + kx - 1;
                if (ix < 0 || ix >= W) continue;
                acc += wc[ky * 3 + kx] * xc[iy * W + ix];
            }
        }
    }
    if (o >= 18) acc = 1.0f / (1.0f + expf(-acc));
    out[idx] = acc;
}

// ---------------------------------------------------------------------------
// Bilinear resize (align_corners-style linspace, matches resize_ac)
// ---------------------------------------------------------------------------
__global__ void resize_kernel(const float* __restrict__ x, float* __restrict__ out,
                              int H, int W, int Ho, int Wo) {
    int idx = blockIdx.x * blockDim.x + threadIdx.x;
    if (idx >= CCH * Ho * Wo) return;
    int wo = idx % Wo;
    int t  = idx / Wo;
    int ho = t % Ho;
    int c  = t / Ho;
    float ys = (Ho > 1) ? ho * (float)(H - 1) / (float)(Ho - 1) : 0.0f;
    float xs = (Wo > 1) ? wo * (float)(W - 1) / (float)(Wo - 1) : 0.0f;
    int y0 = (int)floorf(ys); int y1 = min(y0 + 1, H - 1);
    int x0 = (int)floorf(xs); int x1 = min(x0 + 1, W - 1);
    float wy = ys - (float)y0, wx = xs - (float)x0;
    const float* xc = x + (size_t)c * H * W;
    float t0 = xc[y0 * W + x0] * (1.0f - wy) + xc[y1 * W + x0] * wy;
    float t1 = xc[y0 * W + x1] * (1.0f - wy) + xc[y1 * W + x1] * wy;
    out[idx] = t0 * (1.0f - wx) + t1 * wx;
}

// ---------------------------------------------------------------------------
// Deformable conv as WMMA GEMM, N-blocked.
// Block = 512 threads = 16 waves.  Block handles NPIX output pixels, all 256
// output channels (wave w -> channels [16w, 16w+16), NTILE=NPIX/16 N-tiles).
// K = 2304 = 256ch x 9.  B-panel (72 x NPIX) is sampled cooperatively into
// LDS per chunk with K-pairs interleaved (layout [k/2][pix][k%2]) so each
// WMMA B operand is one aligned ds_load_b64.  Each A float2 load (weights,
// L2-resident) feeds NTILE WMMAs.
// ---------------------------------------------------------------------------
template <int NPIX>
__global__ __launch_bounds__(512)
void deform_gemm_kernel(const float* __restrict__ x, int Hi, int Wi,
                        const float* __restrict__ off,   // [27, Ho*Wo]
                        const float* __restrict__ wgt,   // [256, 2304]
                        float* __restrict__ out,         // [256, Ho*Wo]
                        int Ho, int Wo, int stride) {
    constexpr int NTILE = NPIX / 16;
    const int HWo = Ho * Wo;
    const int HWi = Hi * Wi;

    __shared__ float Blds[KCH * NPIX];     // [k/2][pix][k%2]
    __shared__ float w4[9 * NPIX * 4];
    __shared__ int   ty0[9 * NPIX], ty1[9 * NPIX], tx0[9 * NPIX], tx1[9 * NPIX];

    const int t = threadIdx.x;
    const int pixBase = blockIdx.x * NPIX;

    // Tap table: 9 kernel points x NPIX pixels (mask & validity folded in)
    for (int tt = t; tt < 9 * NPIX; tt += 512) {
        int kk = tt / NPIX, p = tt % NPIX;
        int pix = pixBase + p;
        int h = pix / Wo, w = pix % Wo;
        float offY = off[(2 * kk) * HWo + pix];
        float offX = off[(2 * kk + 1) * HWo + pix];
        float msk  = off[(18 + kk) * HWo + pix];
        float ys = (float)(kk / 3) + (float)(h * stride - 1) + offY;
        float xs = (float)(kk % 3) + (float)(w * stride - 1) + offX;
        float y0f = floorf(ys), x0f = floorf(xs);
        float wy = ys - y0f, wx = xs - x0f;
        float y1f = y0f + 1.0f, x1f = x0f + 1.0f;
        float Hm = (float)(Hi - 1), Wm = (float)(Wi - 1);
        float vy0 = (y0f >= 0.0f && y0f <= Hm) ? 1.0f : 0.0f;
        float vy1 = (y1f >= 0.0f && y1f <= Hm) ? 1.0f : 0.0f;
        float vx0 = (x0f >= 0.0f && x0f <= Wm) ? 1.0f : 0.0f;
        float vx1 = (x1f >= 0.0f && x1f <= Wm) ? 1.0f : 0.0f;
        w4[tt * 4 + 0] = (1.0f - wy) * (1.0f - wx) * vy0 * vx0 * msk;
        w4[tt * 4 + 1] = (1.0f - wy) * wx          * vy0 * vx1 * msk;
        w4[tt * 4 + 2] = wy * (1.0f - wx)          * vy1 * vx0 * msk;
        w4[tt * 4 + 3] = wy * wx                   * vy1 * vx1 * msk;
        ty0[tt] = (int)fminf(fmaxf(y0f, 0.0f), Hm);
        ty1[tt] = (int)fminf(fmaxf(y1f, 0.0f), Hm);
        tx0[tt] = (int)fminf(fmaxf(x0f, 0.0f), Wm);
        tx1[tt] = (int)fminf(fmaxf(x1f, 0.0f), Wm);
    }

    const int lane  = t & 31;
    const int wave  = t >> 5;        // M tile (0..15)
    const int mrow  = lane & 15;     // A row / B col / D col
    const int khalf = lane >> 4;     // selects K pair {0,1} vs {2,3}

    v8f acc[NTILE];
    #pragma unroll
    for (int j = 0; j < NTILE; ++j) acc[j] = (v8f){};

    const float* wbase = wgt + (size_t)(wave * 16 + mrow) * 2304 + 2 * khalf;

    for (int ch = 0; ch < CCH / 8; ++ch) {
        __syncthreads();   // protect Blds reuse across chunks
        int cbase = ch * 8;
        for (int idx = t; idx < KCH * NPIX; idx += 512) {
            int row = idx / NPIX, p = idx % NPIX;
            int c  = cbase + row / 9;
            int kk = row - (row / 9) * 9;
            int tp = kk * NPIX + p;
            const float* xc = x + (size_t)c * HWi;
            float v = w4[tp * 4 + 0] * xc[ty0[tp] * Wi + tx0[tp]]
                    + w4[tp * 4 + 1] * xc[ty0[tp] * Wi + tx1[tp]]
                    + w4[tp * 4 + 2] * xc[ty1[tp] * Wi + tx0[tp]]
                    + w4[tp * 4 + 3] * xc[ty1[tp] * Wi + tx1[tp]];
            // K-pair interleaved store: [row/2][p][row%2]
            Blds[((row >> 1) * NPIX + p) * 2 + (row & 1)] = v;
        }
        __syncthreads();
        int kg = ch * KCH;
        #pragma unroll
        for (int s = 0; s < KCH / 4; ++s) {
            v2f a = *(const v2f*)(wbase + kg + s * 4);
            // B K-rows {s*4+2*khalf, s*4+2*khalf+1} are one contiguous b64
            int rbase = ((s * 2 + khalf) * NPIX + mrow) * 2;
            #pragma unroll
            for (int j = 0; j < NTILE; ++j) {
                v2f b = *(const v2f*)&Blds[rbase + j * 32];
                acc[j] = __builtin_amdgcn_wmma_f32_16x16x4_f32(
                    false, a, false, b, (short)0, acc[j], false, false);
            }
        }
    }

    // D: VGPR v -> M = v (lanes 0-15) / v+8 (lanes 16-31); N = lane&15
    int n    = lane & 15;
    int moff = (lane >> 4) * 8;
    #pragma unroll
    for (int j = 0; j < NTILE; ++j) {
        int pix = pixBase + j * 16 + n;
        #pragma unroll
        for (int v = 0; v < 8; ++v) {
            out[(size_t)(wave * 16 + moff + v) * HWo + pix] = acc[j][v];
        }
    }
}

// ---------------------------------------------------------------------------
// Per-channel sum & sum-of-squares (block per channel)
// ---------------------------------------------------------------------------
__global__ void chan_reduce_kernel(const float* __restrict__ x,
                                   float* __restrict__ sums,
                                   float* __restrict__ sumsq, int HW) {
    __shared__ float s1[256], s2[256];
    int c = blockIdx.x;
    const float* xc = x + (size_t)c * HW;
    float a = 0.0f, b = 0.0f;
    for (int i = threadIdx.x; i < HW; i += 256) { float v = xc[i]; a += v; b += v * v; }
    s1[threadIdx.x] = a; s2[threadIdx.x] = b;
    __syncthreads();
    for (int o = 128; o > 0; o >>= 1) {
        if (threadIdx.x < (unsigned)o) {
            s1[threadIdx.x] += s1[threadIdx.x + o];
            s2[threadIdx.x] += s2[threadIdx.x + o];
        }
        __syncthreads();
    }
    if (threadIdx.x == 0) { sums[c] = s1[0]; sumsq[c] = s2[0]; }
}

// ---------------------------------------------------------------------------
// GroupNorm stats + scale-attention scalar (one block of 256 threads).
// gpar: [0..15]=group mean, [16..31]=group rstd, [32]=attention scalar s.
// Channel mean of the normalized tensor computed analytically from sums.
// ---------------------------------------------------------------------------
__global__ void gn_stats_kernel(const float* __restrict__ sums,
                                const float* __restrict__ sumsq,
                                const float* __restrict__ gng,
                                const float* __restrict__ gnb,
                                const float* __restrict__ saw,
                                const float* __restrict__ sab,
                                float* __restrict__ gpar, int HW) {
    __shared__ float gm[16], grs[16];
    __shared__ float red[256];
    int c = threadIdx.x;
    if (c < 16) {
        float ms = 0.0f, ss = 0.0f;
        for (int j = 0; j < 16; ++j) { ms += sums[c * 16 + j]; ss += sumsq[c * 16 + j]; }
        float n = (float)(16 * HW);
        float m = ms / n;
        float var = ss / n - m * m;
        gm[c] = m; grs[c] = rsqrtf(var + 1e-5f);
        gpar[c] = m; gpar[16 + c] = grs[c];
    }
    __syncthreads();
    int g = c >> 4;
    float cm = sums[c] / (float)HW;
    float ym = (cm - gm[g]) * grs[g] * gng[c] + gnb[c];
    red[c] = saw[c] * ym;
    __syncthreads();
    for (int o = 128; o > 0; o >>= 1) {
        if (c < o) red[c] += red[c + o];
        __syncthreads();
    }
    if (c == 0) {
        float v = fmaxf(red[0] + sab[0], 0.0f);
        gpar[32] = hsig(v);
    }
}

__global__ void gn_apply_kernel(float* __restrict__ x,
                                const float* __restrict__ gng,
                                const float* __restrict__ gnb,
                                const float* __restrict__ gpar, int HW) {
    int idx = blockIdx.x * blockDim.x + threadIdx.x;
    if (idx >= CCH * HW) return;
    int c = idx / HW;
    int g = c >> 4;
    x[idx] = ((x[idx] - gpar[g]) * gpar[16 + g] * gng[c] + gnb[c]) * gpar[32];
}

__global__ void avg_kernel(float* __restrict__ b0, const float* __restrict__ b1,
                           const float* __restrict__ b2, int nb, int HW) {
    int idx = blockIdx.x * blockDim.x + threadIdx.x;
    if (idx >= CCH * HW) return;
    float v = b0[idx] + b1[idx] + (nb == 3 ? b2[idx] : 0.0f);
    b0[idx] = v / (float)nb;
}

// ---------------------------------------------------------------------------
// DyReLU coefficients: 1 block, 1024 threads. coef[k] in [-1,1].
// ---------------------------------------------------------------------------
__global__ void dy_coef_kernel(const float* __restrict__ sums, int HW,
                               const float* __restrict__ l1w,
                               const float* __restrict__ l1b,
                               const float* __restrict__ l2w,
                               const float* __restrict__ l2b,
                               float* __restrict__ coef) {
    __shared__ float h[RDIM];
    __shared__ float m[CCH];
    int t = threadIdx.x;
    if (t < CCH) m[t] = sums[t] / (float)HW;
    __syncthreads();
    if (t < RDIM) {
        float a = l1b[t];
        const float* wr = l1w + (size_t)t * CCH;
        for (int c = 0; c < CCH; ++c) a += m[c] * wr[c];
        h[t] = fmaxf(a, 0.0f);
    }
    __syncthreads();
    float z = l2b[t];
    const float* wr = l2w + (size_t)t * RDIM;
    for (int j = 0; j < RDIM; ++j) z += h[j] * wr[j];
    coef[t] = (hsig(z) - 0.5f) * 2.0f;
}

__global__ void dy_apply_kernel(const float* __restrict__ f,
                                const float* __restrict__ coef,
                                float* __restrict__ out, int HW) {
    int idx = blockIdx.x * blockDim.x + threadIdx.x;
    if (idx >= CCH * HW) return;
    int c = idx / HW;
    float a1  = coef[c] + 1.0f;
    float a2  = coef[CCH + c];
    float bb1 = coef[2 * CCH + c];
    float bb2 = coef[3 * CCH + c];
    float x = f[idx];
    out[idx] = fmaxf(a1 * x + bb1, a2 * x + bb2);
}

// ---------------------------------------------------------------------------
extern "C" void kernel_launch(void* const* d_in, const int* in_sizes, int n_in,
                              void* d_out, int out_size, void* d_ws, size_t ws_size,
                              hipStream_t stream) {
    const float* x0   = (const float*)d_in[0];
    const float* x1   = (const float*)d_in[1];
    const float* x2   = (const float*)d_in[2];
    const float* offw = (const float*)d_in[3];
    const float* offb = (const float*)d_in[4];
    const float* wlow = (const float*)d_in[5];
    const float* wmid = (const float*)d_in[6];
    const float* whigh= (const float*)d_in[7];
    const float* gng  = (const float*)d_in[8];
    const float* gnb  = (const float*)d_in[9];
    const float* saw  = (const float*)d_in[10];
    const float* sab  = (const float*)d_in[11];
    const float* l1w  = (const float*)d_in[12];
    const float* l1b  = (const float*)d_in[13];
    const float* l2w  = (const float*)d_in[14];
    const float* l2b  = (const float*)d_in[15];

    float* ws = (float*)d_ws;
    size_t o = 0;
    float* off_buf = ws + o; o += (size_t)27 * 4096;
    float* hi_buf  = ws + o; o += (size_t)CCH * 4096;
    float* bb[3];
    for (int i = 0; i < 3; ++i) { bb[i] = ws + o; o += (size_t)CCH * 4096; }
    float* l0out = ws + o; o += (size_t)CCH * (4096 + 1024 + 256);
    float* sums  = ws + o; o += 256;
    float* sumsq = ws + o; o += 256;
    float* gpar  = ws + o; o += 64;
    float* coef  = ws + o; o += 1024;

    const int Hs[3] = {64, 32, 16};
    float* fout = (float*)d_out;
    float* l0seg[3]  = { l0out, l0out + (size_t)CCH * 4096,
                         l0out + (size_t)CCH * (4096 + 1024) };
    float* outseg[3] = { fout, fout + (size_t)CCH * 4096,
                         fout + (size_t)CCH * (4096 + 1024) };

    float* off_buf_p = off_buf;
    auto launch_deform = [&](const float* xin, int Hi, const float* wgtp,
                             float* dst, int Ho, int stride) {
        int HW = Ho * Ho;
        if (HW >= 4096) {
            deform_gemm_kernel<64><<<HW / 64, 512, 0, stream>>>(
                xin, Hi, Hi, off_buf_p, wgtp, dst, Ho, Ho, stride);
        } else if (HW >= 1024) {
            deform_gemm_kernel<32><<<HW / 32, 512, 0, stream>>>(
                xin, Hi, Hi, off_buf_p, wgtp, dst, Ho, Ho, stride);
        } else {
            deform_gemm_kernel<16><<<HW / 16, 512, 0, stream>>>(
                xin, Hi, Hi, off_buf_p, wgtp, dst, Ho, Ho, stride);
        }
    };

    for (int l = 0; l < 2; ++l) {
        const float* xs[3];
        float* outp[3];
        xs[0] = (l == 0) ? x0 : l0seg[0];
        xs[1] = (l == 0) ? x1 : l0seg[1];
        xs[2] = (l == 0) ? x2 : l0seg[2];
        for (int i = 0; i < 3; ++i) outp[i] = (l == 0) ? l0seg[i] : outseg[i];

        const float* offw_l = offw + (size_t)l * 27 * CCH * 9;
        const float* offb_l = offb + (size_t)l * 27;
        const float* wlow_l = wlow + (size_t)l * CCH * 2304;
        const float* wmid_l = wmid + (size_t)l * CCH * 2304;
        const float* whigh_l= whigh+ (size_t)l * CCH * 2304;
        const float* gng_l  = gng  + (size_t)l * CCH;
        const float* gnb_l  = gnb  + (size_t)l * CCH;
        const float* saw_l  = saw  + (size_t)l * CCH;
        const float* sab_l  = sab  + (size_t)l;
        const float* l1w_l  = l1w  + (size_t)l * RDIM * CCH;
        const float* l1b_l  = l1b  + (size_t)l * RDIM;
        const float* l2w_l  = l2w  + (size_t)l * 4 * CCH * RDIM;
        const float* l2b_l  = l2b  + (size_t)l * 4 * CCH;

        for (int i = 0; i < 3; ++i) {
            int Ho = Hs[i];
            int HW = Ho * Ho;
            int gE = (CCH * HW + 255) / 256;

            off_conv_kernel<<<(27 * HW + 255) / 256, 256, 0, stream>>>(
                xs[i], offw_l, offb_l, off_buf, Ho, Ho);

            int nb = 0;
            // mid branch
            launch_deform(xs[i], Ho, wmid_l, bb[nb], Ho, 1);
            chan_reduce_kernel<<<256, 256, 0, stream>>>(bb[nb], sums, sumsq, HW);
            gn_stats_kernel<<<1, 256, 0, stream>>>(sums, sumsq, gng_l, gnb_l, saw_l, sab_l, gpar, HW);
            gn_apply_kernel<<<gE, 256, 0, stream>>>(bb[nb], gng_l, gnb_l, gpar, HW);
            ++nb;

            if (i > 0) {  // low branch: sample previous (finer) level with stride 2
                launch_deform(xs[i - 1], Hs[i - 1], wlow_l, bb[nb], Ho, 2);
                chan_reduce_kernel<<<256, 256, 0, stream>>>(bb[nb], sums, sumsq, HW);
                gn_stats_kernel<<<1, 256, 0, stream>>>(sums, sumsq, gng_l, gnb_l, saw_l, sab_l, gpar, HW);
                gn_apply_kernel<<<gE, 256, 0, stream>>>(bb[nb], gng_l, gnb_l, gpar, HW);
                ++nb;
            }
            if (i < 2) {  // high branch: upsample next (coarser) level, stride 1
                int Hin = Hs[i + 1];
                resize_kernel<<<gE, 256, 0, stream>>>(xs[i + 1], hi_buf, Hin, Hin, Ho, Ho);
                launch_deform(hi_buf, Ho, whigh_l, bb[nb], Ho, 1);
                chan_reduce_kernel<<<256, 256, 0, stream>>>(bb[nb], sums, sumsq, HW);
                gn_stats_kernel<<<1, 256, 0, stream>>>(sums, sumsq, gng_l, gnb_l, saw_l, sab_l, gpar, HW);
                gn_apply_kernel<<<gE, 256, 0, stream>>>(bb[nb], gng_l, gnb_l, gpar, HW);
                ++nb;
            }

            if (nb > 1)
                avg_kernel<<<gE, 256, 0, stream>>>(bb[0], bb[1], bb[2], nb, HW);

            chan_reduce_kernel<<<256, 256, 0, stream>>>(bb[0], sums, sumsq, HW);
            dy_coef_kernel<<<1, 1024, 0, stream>>>(sums, HW, l1w_l, l1b_l, l2w_l, l2b_l, coef);
            dy_apply_kernel<<<gE, 256, 0, stream>>>(bb[0], coef, outp[i], HW);
        }
    }
}